// YOLOv2_4569845203302
// MI455X (gfx1250) — compile-verified
//
#include <hip/hip_runtime.h>
#include <math.h>

// YOLOv2 loss for gfx1250 (MI455X).
// Memory/latency-bound (~31MB traffic, ~0.4 GFLOP) -> optimize data movement:
// async-to-LDS staging of GT boxes, coalesced plane loads, LDS tree reductions.

#define YW   52
#define YH   52
#define YHW  (YW * YH)          // 2704 cells
#define YA   5                  // anchors
#define YM   32                 // gt boxes per image
#define YC   20                 // classes
#define YCPA (5 + YC)           // channels per anchor
#define YCT  (YA * YCPA)        // 125 channels total
#define TPB  256                // 8 wave32 per block

// Types matching the builtin's declared signature:
//   param0: __device__ (AS1) pointer to 16B int vector
//   param1: LDS (AS3) pointer to 16B int vector
typedef int v4i __attribute__((vector_size(16)));
typedef __attribute__((address_space(1))) v4i gv4i;
typedef __attribute__((address_space(3))) v4i lv4i;

__global__ __launch_bounds__(TPB) void yolo_part_kernel(
    const float* __restrict__ x,        // (N, 125, 52, 52)
    const float* __restrict__ anchors,  // (5, 2)
    const float* __restrict__ gt_boxes, // (N, 32, 4)
    const int*   __restrict__ gt_labels,// (N, 32)
    const int*   __restrict__ img_h,    // scalar
    const int*   __restrict__ img_w,    // scalar
    float* __restrict__ partial)        // (gridDim.y*gridDim.x, 3)
{
    __shared__ float sg[YM][4];
    __shared__ float sarea[YM];
    __shared__ int   slab[YM];
    __shared__ float red[3][TPB];

    const int tid  = threadIdx.x;
    const int n    = blockIdx.y;
    const int cell = blockIdx.x * TPB + tid;

    // ---- Stage this image's GT boxes (512B) into LDS -----------------------
    if (tid < YM) {
        slab[tid] = gt_labels[n * YM + tid];
#if defined(__gfx1250__) && __has_builtin(__builtin_amdgcn_global_load_async_to_lds_b128)
        // CDNA5 async DMA path: 32 lanes x 16B, tracked by ASYNCcnt.
        __builtin_amdgcn_global_load_async_to_lds_b128(
            (gv4i*)(gt_boxes + (size_t)(n * YM + tid) * 4),
            (lv4i*)(&sg[tid][0]),
            0, 0);
#if __has_builtin(__builtin_amdgcn_s_wait_asynccnt)
        __builtin_amdgcn_s_wait_asynccnt(0);
#else
        asm volatile("s_wait_asynccnt 0" ::: "memory");
#endif
#else
        const float4 gb = ((const float4*)gt_boxes)[n * YM + tid];
        sg[tid][0] = gb.x; sg[tid][1] = gb.y; sg[tid][2] = gb.z; sg[tid][3] = gb.w;
#endif
    }
    __syncthreads();
    if (tid < YM)
        sarea[tid] = (sg[tid][2] - sg[tid][0]) * (sg[tid][3] - sg[tid][1]);
    __syncthreads();

    const float sx = (float)img_w[0] / (float)YW;   // 32.0
    const float sy = (float)img_h[0] / (float)YH;   // 32.0

    float obj = 0.0f, bbx = 0.0f, clf = 0.0f;

    if (cell < YHW) {
        const int hh = cell / YW;
        const int ww = cell - hh * YW;
        const float cx = (float)ww * sx;
        const float cy = (float)hh * sy;
        const float* xb = x + (size_t)n * YCT * YHW + cell;

        float best_m = -1.0f;           // max IoU over (a,g)
        int   best_a = 0, best_g = 0;
        float sbx = 0.f, sby = 0.f, sx2 = 0.f, sy2 = 0.f, sel_o = 0.f;
        float max_o = 0.0f;             // max objectness over anchors

#pragma unroll
        for (int a = 0; a < YA; ++a) {
            const float* p = xb + (size_t)(a * YCPA) * YHW;
            const float t0 = p[0 * YHW];
            const float t1 = p[1 * YHW];
            const float t2 = p[2 * YHW];
            const float t3 = p[3 * YHW];
            const float t4 = p[4 * YHW];

            const float bx = 1.0f / (1.0f + __expf(-t0)) + cx;
            const float by = 1.0f / (1.0f + __expf(-t1)) + cy;
            const float bw = sx * anchors[2 * a + 0] * __expf(t2);
            const float bh = sy * anchors[2 * a + 1] * __expf(t3);
            const float x2 = bx + bw;
            const float y2 = by + bh;
            const float ap = bw * bh;
            const float o  = 1.0f / (1.0f + __expf(-t4));
            max_o = fmaxf(max_o, o);

            // per-anchor max IoU + first-argmax over 32 GT boxes
            float amax = -1.0f; int ag = 0;
            for (int g = 0; g < YM; ++g) {
                const float ltx = fmaxf(bx, sg[g][0]);
                const float lty = fmaxf(by, sg[g][1]);
                const float rbx = fminf(x2, sg[g][2]);
                const float rby = fminf(y2, sg[g][3]);
                const float wi = fmaxf(rbx - ltx, 0.0f);
                const float hi = fmaxf(rby - lty, 0.0f);
                const float inter = wi * hi;
                const float iou = inter / (ap + sarea[g] - inter);
                if (iou > amax) { amax = iou; ag = g; }   // strict > == jnp first-argmax
            }
            if (amax > best_m) {
                best_m = amax; best_a = a; best_g = ag;
                sbx = bx; sby = by; sx2 = x2; sy2 = y2; sel_o = o;
            }
        }

        if (best_m > 0.0f) {            // cell_sel: m == 1
            const float d = sel_o - best_m;
            obj = d * d;

            const float gx1 = sg[best_g][0], gy1 = sg[best_g][1];
            const float gx2 = sg[best_g][2], gy2 = sg[best_g][3];
            const float dx = sbx - gx1;
            const float dy = sby - gy1;
            const float dw = sqrtf(sx2) - sqrtf(gx2);
            const float dh = sqrtf(sy2) - sqrtf(gy2);
            bbx = dx * dx + dy * dy + dw * dw + dh * dh;

            // cross-entropy on the best anchor's 20 class logits
            const float* ps = xb + (size_t)(best_a * YCPA + 5) * YHW;
            float s[YC];
            float mx = -3.402823466e38f;
#pragma unroll
            for (int k = 0; k < YC; ++k) {
                s[k] = ps[(size_t)k * YHW];
                mx = fmaxf(mx, s[k]);
            }
            float se = 0.0f;
#pragma unroll
            for (int k = 0; k < YC; ++k) se += __expf(s[k] - mx);
            const float lse = mx + __logf(se);
            clf = lse - s[slab[best_g]];
        } else {                        // m == 0: only 0.5 * max_o^2 term
            obj = 0.5f * max_o * max_o;
        }
    }

    // ---- deterministic block reduction ------------------------------------
    red[0][tid] = obj; red[1][tid] = bbx; red[2][tid] = clf;
    __syncthreads();
    for (int st = TPB / 2; st > 0; st >>= 1) {
        if (tid < st) {
            red[0][tid] += red[0][tid + st];
            red[1][tid] += red[1][tid + st];
            red[2][tid] += red[2][tid + st];
        }
        __syncthreads();
    }
    if (tid == 0) {
        const int b = blockIdx.y * gridDim.x + blockIdx.x;
        partial[b * 3 + 0] = red[0][0];
        partial[b * 3 + 1] = red[1][0];
        partial[b * 3 + 2] = red[2][0];
    }
}

__global__ __launch_bounds__(TPB) void yolo_final_kernel(
    const float* __restrict__ partial, int nblocks, float* __restrict__ out)
{
    __shared__ float red[3][TPB];
    const int tid = threadIdx.x;
    float a = 0.0f, b = 0.0f, c = 0.0f;
    for (int j = tid; j < nblocks; j += TPB) {   // fixed order -> deterministic
        a += partial[j * 3 + 0];
        b += partial[j * 3 + 1];
        c += partial[j * 3 + 2];
    }
    red[0][tid] = a; red[1][tid] = b; red[2][tid] = c;
    __syncthreads();
    for (int st = TPB / 2; st > 0; st >>= 1) {
        if (tid < st) {
            red[0][tid] += red[0][tid + st];
            red[1][tid] += red[1][tid + st];
            red[2][tid] += red[2][tid + st];
        }
        __syncthreads();
    }
    if (tid == 0) {
        out[0] = red[0][0];   // objectness_loss
        out[1] = red[1][0];   // bbox_loss
        out[2] = red[2][0];   // clf_loss
    }
}

extern "C" void kernel_launch(void* const* d_in, const int* in_sizes, int n_in,
                              void* d_out, int out_size, void* d_ws, size_t ws_size,
                              hipStream_t stream) {
    const float* x         = (const float*)d_in[0];
    const float* anchors   = (const float*)d_in[1];
    const float* gt_boxes  = (const float*)d_in[2];
    const int*   gt_labels = (const int*)d_in[3];
    const int*   img_h     = (const int*)d_in[4];
    const int*   img_w     = (const int*)d_in[5];
    float* out     = (float*)d_out;
    float* partial = (float*)d_ws;

    const int N = in_sizes[0] / (YCT * YHW);     // 64
    dim3 grid((YHW + TPB - 1) / TPB, N);         // (11, 64)

    yolo_part_kernel<<<grid, TPB, 0, stream>>>(x, anchors, gt_boxes, gt_labels,
                                               img_h, img_w, partial);
    const int nblocks = grid.x * grid.y;
    yolo_final_kernel<<<1, TPB, 0, stream>>>(partial, nblocks, out);
}